// GHMC_67929202754192
// MI455X (gfx1250) — compile-verified
//
#include <hip/hip_runtime.h>

// GHM-C loss, single-pass streaming formulation:
//   loss = ( sum_b  bce_sum[b] / counts[b] ) / n_nonempty
//
// Per-bin accumulator packed into one u64:  [63:42] = count, [41:0] = sum of
// round(bce * 2^20).  bce in [0.31, 1.32] (always positive) so each quantum
// fits in 21 bits; per-wave-bin totals stay far below the field boundaries,
// so a single integer ds_add_u64 per element carries both count and sum.
// Integer-only atomics -> bit-exact deterministic result.

#define BINS 10
#define TPB 256
#define NWAVES (TPB / 32)     // wave32 on gfx1250 -> 8 waves per 256-thread block
#define GRID_BLOCKS 1024

#define CNT_SHIFT 42
#define SUM_MASK ((1ull << CNT_SHIFT) - 1ull)
#define FIX_SCALE 1048576.0f          // 2^20
#define INV_FIX_SCALE (1.0 / 1048576.0)

struct GhmcWs {
  unsigned long long sumfix[BINS];
  unsigned int counts[BINS];
};

__global__ void ghmc_init_kernel(GhmcWs* __restrict__ ws) {
  int t = threadIdx.x;
  if (t < BINS) {
    ws->sumfix[t] = 0ull;
    ws->counts[t] = 0u;
  }
}

__device__ __forceinline__ unsigned long long ghmc_pack(float p, float tf) {
  const float L2E = 1.4426950408889634f;  // log2(e)
  const float LN2 = 0.6931471805599453f;  // ln(2)
  // z = sigmoid(p) via hardware v_exp_f32 (2^x) + v_rcp_f32
  float z = __builtin_amdgcn_rcpf(1.0f + __builtin_amdgcn_exp2f(-p * L2E));
  // softplus(z) = ln(1 + e^z); z in (0,1) so no overflow concerns
  float sp = __builtin_amdgcn_logf(1.0f + __builtin_amdgcn_exp2f(z * L2E)) * LN2;
  float bce = sp - tf * z;                 // in (0.31, 1.32) -> always positive
  float g = fabsf(z - tf);                 // in [0,1]
  int b = (int)(g * 10.0f);                // floor, g >= 0
  b = b > (BINS - 1) ? (BINS - 1) : b;
  unsigned int q = (unsigned int)(bce * FIX_SCALE + 0.5f);   // < 2^21
  unsigned long long pack = (1ull << CNT_SHIFT) | (unsigned long long)q;
  // return bin in low bits of a second value via pair; caller needs (b, pack)
  // -> fold bin into the return by having caller recompute? keep it simple:
  return pack | ((unsigned long long)b << 62);  // unused; see note below
}

__launch_bounds__(TPB)
__global__ void ghmc_hist_kernel(const float* __restrict__ pred,
                                 const int* __restrict__ tgt,
                                 GhmcWs* __restrict__ ws,
                                 int n) {
  // Per-wave replicated packed histograms in LDS to cut ds_add contention.
  __shared__ unsigned long long lpack[NWAVES][BINS];

  const int tid = threadIdx.x;
  const int wave = tid >> 5;  // wave32

  for (int i = tid; i < NWAVES * BINS; i += TPB) {
    (&lpack[0][0])[i] = 0ull;
  }
  __syncthreads();

  const float L2E = 1.4426950408889634f;  // log2(e)
  const float LN2 = 0.6931471805599453f;  // ln(2)

  const int n4 = n >> 2;
  const int stride = GRID_BLOCKS * TPB;
  const float4* __restrict__ pred4 = (const float4*)pred;
  const int4* __restrict__ tgt4 = (const int4*)tgt;

  for (int i = blockIdx.x * TPB + tid; i < n4; i += stride) {
    float4 p4 = pred4[i];
    int4 t4 = tgt4[i];
    // gfx1250 global_prefetch_b8 for the next grid-stride tile (speculative,
    // OOB translations are silently dropped).
    __builtin_prefetch((const void*)(pred4 + i + stride), 0, 1);
    __builtin_prefetch((const void*)(tgt4 + i + stride), 0, 1);

    float pv[4] = {p4.x, p4.y, p4.z, p4.w};
    int ti[4] = {t4.x, t4.y, t4.z, t4.w};
#pragma unroll
    for (int k = 0; k < 4; ++k) {
      float z = __builtin_amdgcn_rcpf(1.0f + __builtin_amdgcn_exp2f(-pv[k] * L2E));
      float tf = (float)ti[k];
      float sp = __builtin_amdgcn_logf(1.0f + __builtin_amdgcn_exp2f(z * L2E)) * LN2;
      float bce = sp - tf * z;             // > 0 always
      float g = fabsf(z - tf);             // in [0,1]
      int b = (int)(g * 10.0f);
      b = b > (BINS - 1) ? (BINS - 1) : b;
      unsigned int q = (unsigned int)(bce * FIX_SCALE + 0.5f);   // < 2^21
      unsigned long long pk = (1ull << CNT_SHIFT) | (unsigned long long)q;
      atomicAdd(&lpack[wave][b], pk);      // single ds_add_u64: count + sum
    }
  }

  // Scalar tail (n not multiple of 4 — not the case here, but keep it correct).
  for (int i = (n4 << 2) + blockIdx.x * TPB + tid; i < n; i += stride) {
    float p = pred[i];
    float tf = (float)tgt[i];
    float z = __builtin_amdgcn_rcpf(1.0f + __builtin_amdgcn_exp2f(-p * L2E));
    float sp = __builtin_amdgcn_logf(1.0f + __builtin_amdgcn_exp2f(z * L2E)) * LN2;
    float bce = sp - tf * z;
    float g = fabsf(z - tf);
    int b = (int)(g * 10.0f);
    b = b > (BINS - 1) ? (BINS - 1) : b;
    unsigned int q = (unsigned int)(bce * FIX_SCALE + 0.5f);
    unsigned long long pk = (1ull << CNT_SHIFT) | (unsigned long long)q;
    atomicAdd(&lpack[wave][b], pk);
  }

  __syncthreads();

  // Fold the 8 wave histograms; one u64 + one u32 global atomic per bin per
  // block (10,240 each total -> negligible, and fully deterministic).
  if (tid < BINS) {
    unsigned long long v = 0ull;
#pragma unroll
    for (int w = 0; w < NWAVES; ++w) {
      v += lpack[w][tid];                  // fields cannot carry: sum < 2^42
    }
    unsigned int c = (unsigned int)(v >> CNT_SHIFT);
    unsigned long long sfix = v & SUM_MASK;
    if (c > 0u) {
      atomicAdd(&ws->sumfix[tid], sfix);   // global_atomic_add_u64
      atomicAdd(&ws->counts[tid], c);      // global_atomic_add_u32
    }
  }
}

__global__ void ghmc_final_kernel(const GhmcWs* __restrict__ ws,
                                  float* __restrict__ out) {
  if (threadIdx.x == 0 && blockIdx.x == 0) {
    int nn = 0;
    double acc = 0.0;
#pragma unroll
    for (int b = 0; b < BINS; ++b) {
      unsigned int c = ws->counts[b];
      if (c > 0u) {
        nn++;
        acc += ((double)ws->sumfix[b] * INV_FIX_SCALE) / (double)c;
      }
    }
    double d = (nn > 0) ? (double)nn : 1.0;
    out[0] = (float)(acc / d);
  }
}

extern "C" void kernel_launch(void* const* d_in, const int* in_sizes, int n_in,
                              void* d_out, int out_size, void* d_ws, size_t ws_size,
                              hipStream_t stream) {
  const float* pred = (const float*)d_in[0];
  const int* tgt = (const int*)d_in[1];
  GhmcWs* ws = (GhmcWs*)d_ws;        // 120 bytes of scratch used
  float* out = (float*)d_out;
  const int n = in_sizes[0];         // N*C = 20,971,520

  hipLaunchKernelGGL(ghmc_init_kernel, dim3(1), dim3(32), 0, stream, ws);
  hipLaunchKernelGGL(ghmc_hist_kernel, dim3(GRID_BLOCKS), dim3(TPB), 0, stream,
                     pred, tgt, ws, n);
  hipLaunchKernelGGL(ghmc_final_kernel, dim3(1), dim3(32), 0, stream, ws, out);
}